// GNNEncoder_53386443489751
// MI455X (gfx1250) — compile-verified
//
#include <hip/hip_runtime.h>
#include <hip/hip_bf16.h>

typedef __attribute__((ext_vector_type(2))) float v2f;
typedef __attribute__((ext_vector_type(4))) float v4f;
typedef __attribute__((ext_vector_type(8))) float v8f;

#define DIM 64
#define KEEP_PROB 0.7f

// ---------------------------------------------------------------------------
// Kernel 1: acc = all_emb @ W^T + bias  using V_WMMA_F32_16X16X4_F32
//   block = 256 threads (8 waves), each wave -> 16 output rows, all 64 cols.
//   W (16 KB) staged in LDS once per block; per-wave 16x64 A tile staged in
//   LDS with coalesced float4 loads.
// ---------------------------------------------------------------------------
__global__ void __launch_bounds__(256)
gcn_fc_wmma_kernel(const float* __restrict__ emb, const float* __restrict__ W,
                   const float* __restrict__ bias, float* __restrict__ acc,
                   int n_nodes) {
    __shared__ alignas(16) float Wlds[DIM * DIM];      // W row-major [n][k]
    __shared__ alignas(16) float Alds[8][16 * DIM];    // per-wave emb tile

    const int tid  = threadIdx.x;
    const int wave = tid >> 5;
    const int lane = tid & 31;

    // Cooperative load of W: 4096 floats = 1024 float4 over 256 threads
    for (int i = tid; i < (DIM * DIM) / 4; i += 256) {
        ((v4f*)Wlds)[i] = ((const v4f*)W)[i];
    }

    const int row_base = blockIdx.x * 128 + wave * 16;

    // Stage this wave's 16x64 A tile: 256 float4 over 32 lanes, coalesced
    for (int i = lane; i < 256; i += 32) {
        int r  = i >> 4;   // 0..15 (row in tile)
        int c4 = i & 15;   // 0..15 (float4 column)
        int gr = row_base + r;
        v4f v = (gr < n_nodes) ? ((const v4f*)(emb + (size_t)gr * DIM))[c4]
                               : (v4f){0.f, 0.f, 0.f, 0.f};
        ((v4f*)&Alds[wave][0])[r * 16 + c4] = v;
    }
    __syncthreads();

    const float* At = &Alds[wave][0];

    // A fragment addressing: lane = m + 16*(k>=2), vgpr = k%2
    const int m_rel = lane & 15;
    const int k_off = (lane >> 4) * 2;   // 0 or 2
    const int n_lo  = lane & 15;

    v8f c0 = {}, c1 = {}, c2 = {}, c3 = {};

#pragma unroll
    for (int kk = 0; kk < 16; ++kk) {
        const int k = kk * 4 + k_off;
        v2f a  = *(const v2f*)(At + m_rel * DIM + k);
        // B[k][n] = W[n][k]; contiguous float2 along k in LDS row n
        v2f b0 = *(const v2f*)(Wlds + (n_lo +  0) * DIM + k);
        v2f b1 = *(const v2f*)(Wlds + (n_lo + 16) * DIM + k);
        v2f b2 = *(const v2f*)(Wlds + (n_lo + 32) * DIM + k);
        v2f b3 = *(const v2f*)(Wlds + (n_lo + 48) * DIM + k);
        c0 = __builtin_amdgcn_wmma_f32_16x16x4_f32(false, a, false, b0, (short)0, c0, false, false);
        c1 = __builtin_amdgcn_wmma_f32_16x16x4_f32(false, a, false, b1, (short)0, c1, false, false);
        c2 = __builtin_amdgcn_wmma_f32_16x16x4_f32(false, a, false, b2, (short)0, c2, false, false);
        c3 = __builtin_amdgcn_wmma_f32_16x16x4_f32(false, a, false, b3, (short)0, c3, false, false);
    }

    // Bias (broadcast along M, depends only on N)
    const float bb0 = bias[n_lo];
    const float bb1 = bias[n_lo + 16];
    const float bb2 = bias[n_lo + 32];
    const float bb3 = bias[n_lo + 48];

    // C/D layout: vgpr j holds M = j + 8*(lane>=16), N = lane%16
    const int m_hi = (lane >> 4) * 8;
#pragma unroll
    for (int j = 0; j < 8; ++j) {
        int gm = row_base + m_hi + j;
        if (gm < n_nodes) {
            float* outr = acc + (size_t)gm * DIM;
            outr[n_lo +  0] = c0[j] + bb0;
            outr[n_lo + 16] = c1[j] + bb1;
            outr[n_lo + 32] = c2[j] + bb2;
            outr[n_lo + 48] = c3[j] + bb3;
        }
    }
}

// ---------------------------------------------------------------------------
// Kernel 2: one GCN layer SpMM with edge dropout.
//   One wave per edge: per-edge scalars forced uniform via readfirstlane
//   (scalar loads + uniform s_cbranch skip for dropped edges); 32 lanes x
//   float2 = full 64-float row coalesced gather; scatter via inline-asm
//   global_atomic_add_f32 (no-return, DEV scope -> L2 atomic units).
// ---------------------------------------------------------------------------
__global__ void __launch_bounds__(256)
gcn_spmm_kernel(const float* __restrict__ vals, const float* __restrict__ dropu,
                const int* __restrict__ rows, const int* __restrict__ cols,
                const float* __restrict__ src, float* __restrict__ dst,
                int n_edges) {
    const long long tid = (long long)blockIdx.x * blockDim.x + threadIdx.x;
    int e = (int)(tid >> 5);
    if (e >= n_edges) return;
    // e is identical across the 32 lanes of this wave -> force into SGPR
    e = __builtin_amdgcn_readfirstlane(e);
    const int l = threadIdx.x & 31;

    const float u = dropu[e];
    if (u + KEEP_PROB < 1.0f) return;          // dropped edge: whole wave skips
    const float v = vals[e] * (1.0f / KEEP_PROB);

    const int col = cols[e];
    const int row = rows[e];

    const float* sp = src + (size_t)col * DIM + l * 2;
    float*       dp = dst + (size_t)row * DIM + l * 2;
    const float m0 = v * sp[0];
    const float m1 = v * sp[1];

    // Guaranteed single hardware fp-atomic (no CAS-loop expansion).
    // No-return -> STOREcnt; implicit wait-idle at s_endpgm drains it.
    asm volatile("global_atomic_add_f32 %0, %1, off scope:SCOPE_DEV"
                 :: "v"(dp), "v"(m0) : "memory");
    asm volatile("global_atomic_add_f32 %0, %1, off scope:SCOPE_DEV"
                 :: "v"(dp + 1), "v"(m1) : "memory");
}

// ---------------------------------------------------------------------------
// Kernel 3: out = (out + x1) / 3, streaming float4
//   (out already holds fc + x2: gemm wrote fc, layer-1 atomics added x2)
// ---------------------------------------------------------------------------
__global__ void __launch_bounds__(256)
gcn_finalize_kernel(float* __restrict__ out, const float* __restrict__ x1,
                    long long n4) {
    const long long i = (long long)blockIdx.x * blockDim.x + threadIdx.x;
    if (i >= n4) return;
    v4f a = ((const v4f*)out)[i];
    v4f p = ((const v4f*)x1)[i];
    v4f r = (a + p) * (1.0f / 3.0f);
    ((v4f*)out)[i] = r;
}

extern "C" void kernel_launch(void* const* d_in, const int* in_sizes, int n_in,
                              void* d_out, int out_size, void* d_ws, size_t ws_size,
                              hipStream_t stream) {
    const float* emb   = (const float*)d_in[0];   // [N, 64]
    const float* W     = (const float*)d_in[1];   // [64, 64]
    const float* bias  = (const float*)d_in[2];   // [64]
    const float* vals  = (const float*)d_in[3];   // [E]
    const float* dropu = (const float*)d_in[4];   // [2, E]
    const int*   rows  = (const int*)d_in[5];     // [E]
    const int*   cols  = (const int*)d_in[6];     // [E]

    const int N = in_sizes[0] / DIM;
    const int E = in_sizes[3];

    float* out = (float*)d_out;
    float* x1  = (float*)d_ws;                    // [N, 64] layer-0 output

    // Zero the layer-0 atomic accumulator (re-done every call -> deterministic)
    hipMemsetAsync(d_ws, 0, (size_t)N * DIM * sizeof(float), stream);

    // fc(all_emb) -> d_out via WMMA
    {
        dim3 grid((N + 127) / 128), block(256);
        gcn_fc_wmma_kernel<<<grid, block, 0, stream>>>(emb, W, bias, out, N);
    }

    // Layer 0: x1 = A_drop0 @ all_emb (into ws)
    // Layer 1: x2 = A_drop1 @ x1, atomically accumulated straight into d_out
    {
        long long threads = (long long)E * 32;
        dim3 grid((unsigned)((threads + 255) / 256)), block(256);
        gcn_spmm_kernel<<<grid, block, 0, stream>>>(vals, dropu,     rows, cols, emb, x1,  E);
        gcn_spmm_kernel<<<grid, block, 0, stream>>>(vals, dropu + E, rows, cols, x1,  out, E);
    }

    // out = (fc + x2 + x1) / 3
    {
        long long n4 = (long long)N * DIM / 4;
        dim3 grid((unsigned)((n4 + 255) / 256)), block(256);
        gcn_finalize_kernel<<<grid, block, 0, stream>>>(out, x1, n4);
    }
}